// PGCN_59657095741762
// MI455X (gfx1250) — compile-verified
//
#include <hip/hip_runtime.h>
#include <cmath>

#define N_NODES   50000
#define N_EDGES   800000
#define F_IN      128
#define HID       128
#define N_CLASSES 16

typedef __attribute__((ext_vector_type(2))) float v2f;
typedef __attribute__((ext_vector_type(8))) float v8f;

// ---------------- degree / normalization ----------------
__global__ void k_zero(float* __restrict__ p, int n) {
  int t = blockIdx.x * blockDim.x + threadIdx.x;
  if (t < n) p[t] = 0.0f;
}

__global__ void k_deg(const int* __restrict__ dst, float* __restrict__ deg, int e) {
  int t = blockIdx.x * blockDim.x + threadIdx.x;
  if (t < e) atomicAdd(&deg[dst[t]], 1.0f);
}

__global__ void k_dinv(float* __restrict__ p, int n) {
  int t = blockIdx.x * blockDim.x + threadIdx.x;
  if (t < n) p[t] = rsqrtf(p[t] + 2.0f);   // improved=True -> +2 self-loop weight
}

// ---------------- dense transform: H = X @ W via f32 WMMA ----------------
// One wave computes a 16 x (16*NT) row-block of H, so each A load feeds NT
// wmmas. Whole W (K x N) is staged in LDS once per block (row stride padded
// so the half-wave split rows k / k+2 hit disjoint banks). K fixed at 128.
template <int NT>
__global__ void __launch_bounds__(256)
k_gemm_wmma(const float* __restrict__ X, const float* __restrict__ W,
            float* __restrict__ H, int M) {
  constexpr int K = 128;
  constexpr int N = 16 * NT;
  constexpr int LDSW = N + 8;              // 2*LDSW % 64 == 16 words -> no bank conflict
  __shared__ float sW[K * LDSW];

  // cooperative stage of W into LDS (N constexpr -> div/mod are shifts)
  for (int i = threadIdx.x; i < K * N; i += blockDim.x) {
    int k = i / N, n = i - k * N;
    sW[k * LDSW + n] = W[i];
  }
  __syncthreads();                         // only barrier; uniform exit below is safe

  const int wave = blockIdx.x * (blockDim.x >> 5) + (threadIdx.x >> 5);
  const int lane = threadIdx.x & 31;
  const int m0 = wave << 4;
  if (m0 >= M) return;                     // wave-uniform: EXEC all-ones for WMMA
  const int lm = lane & 15;
  const int g  = lane >> 4;                // half-wave group
  const int hi = g << 1;                   // K sub-offset 0 or 2 (ISA A/B f32 layout)

  v8f acc[NT];
#pragma unroll
  for (int t = 0; t < NT; ++t) acc[t] = (v8f){};

  const float* xr = X + (size_t)(m0 + lm) * K + hi;
  const float* wr = sW + hi * LDSW + lm;
#pragma unroll 2
  for (int k0 = 0; k0 < K; k0 += 4) {
    v2f a;
    a.x = xr[k0];                          // A: V0 = K 2g, V1 = K 2g+1
    a.y = xr[k0 + 1];
    const float* wk = wr + k0 * LDSW;
#pragma unroll
    for (int t = 0; t < NT; ++t) {
      v2f b;
      b.x = wk[t * 16];                    // B: V0 = row 2g, V1 = row 2g+1
      b.y = wk[t * 16 + LDSW];
      acc[t] = __builtin_amdgcn_wmma_f32_16x16x4_f32(
          false, a, false, b, (short)0, acc[t], false, false);
    }
  }

  // C/D layout: VGPR r -> row m0+r (lanes 0-15) / m0+r+8 (lanes 16-31)
  const int rbase = g << 3;
#pragma unroll
  for (int t = 0; t < NT; ++t)
#pragma unroll
    for (int r = 0; r < 8; ++r)
      H[(size_t)(m0 + rbase + r) * N + t * 16 + lm] = acc[t][r];
}

// ---------------- agg = (2 * dinv^2) * h  (self-loop term, also init) ----------------
template <int F>
__global__ void k_selfinit(const float* __restrict__ h, const float* __restrict__ dinv,
                           float* __restrict__ agg, int n) {
  constexpr int Q = F >> 2;
  int t = blockIdx.x * blockDim.x + threadIdx.x;
  if (t >= n * Q) return;
  int node = t / Q;                        // Q constexpr -> shift
  float c = 2.0f * dinv[node] * dinv[node];
  float4 hv = ((const float4*)h)[t];
  ((float4*)agg)[t] = make_float4(hv.x * c, hv.y * c, hv.z * c, hv.w * c);
}

// ---------------- edge scatter: agg[dst] += dinv[src]*dinv[dst] * h[src] ----------------
template <int F>
__global__ void k_edge(const float* __restrict__ h, const float* __restrict__ dinv,
                       const int* __restrict__ src, const int* __restrict__ dst,
                       float* __restrict__ agg, int E) {
  constexpr int Q = F >> 2;                // float4 chunks per edge
  int t = blockIdx.x * blockDim.x + threadIdx.x;
  if (t >= E * Q) return;                  // 25.6M max, fits int
  int e = t / Q;
  int q = t - e * Q;
  int s = src[e], d = dst[e];
  float c = dinv[s] * dinv[d];
  float4 hv = ((const float4*)(h + (size_t)s * F))[q];
  float* out = agg + (size_t)d * F + (q << 2);
  atomicAdd(out + 0, hv.x * c);
  atomicAdd(out + 1, hv.y * c);
  atomicAdd(out + 2, hv.z * c);
  atomicAdd(out + 3, hv.w * c);
}

// ---------------- out = act(agg + b), float4-vectorized ----------------
template <int F>
__global__ void k_bias_act(const float* __restrict__ agg, const float* __restrict__ b,
                           float* __restrict__ out, int n, int act) {
  constexpr int Q = F >> 2;
  int t = blockIdx.x * blockDim.x + threadIdx.x;
  if (t >= n * Q) return;
  int f4 = t % Q;
  float4 v = ((const float4*)agg)[t];
  float4 bv = ((const float4*)b)[f4];
  v.x += bv.x; v.y += bv.y; v.z += bv.z; v.w += bv.w;
  float4 o;
  if (act) {
    o = make_float4(tanhf(v.x), tanhf(v.y), tanhf(v.z), tanhf(v.w));
  } else {
    o = make_float4(fmaxf(v.x, 0.f), fmaxf(v.y, 0.f), fmaxf(v.z, 0.f), fmaxf(v.w, 0.f));
  }
  ((float4*)out)[t] = o;
}

// ---------------- one GCN layer (K is always 128) ----------------
template <int FOUT>
static void run_layer(const float* X, const float* W, const float* b,
                      float* bufH, float* bufA, float* out,
                      const float* dinv, const int* src, const int* dst,
                      int act, hipStream_t stream) {
  constexpr int NT = FOUT / 16;
  constexpr int Q  = FOUT / 4;
  int waves  = (N_NODES + 15) / 16;                   // 3125
  int blocks = (waves + 7) / 8;                       // 8 waves / 256-thread block
  k_gemm_wmma<NT><<<blocks, 256, 0, stream>>>(X, W, bufH, N_NODES);

  int total4 = N_NODES * Q;
  k_selfinit<FOUT><<<(total4 + 255) / 256, 256, 0, stream>>>(bufH, dinv, bufA, N_NODES);

  int totalE = N_EDGES * Q;
  k_edge<FOUT><<<(totalE + 255) / 256, 256, 0, stream>>>(bufH, dinv, src, dst, bufA, N_EDGES);

  k_bias_act<FOUT><<<(total4 + 255) / 256, 256, 0, stream>>>(bufA, b, out, N_NODES, act);
}

extern "C" void kernel_launch(void* const* d_in, const int* in_sizes, int n_in,
                              void* d_out, int out_size, void* d_ws, size_t ws_size,
                              hipStream_t stream) {
  (void)in_sizes; (void)n_in; (void)out_size; (void)ws_size;

  const float* x  = (const float*)d_in[0];
  const int*   ei = (const int*)d_in[1];          // int32 (JAX x64 disabled)
  const int* src = ei;                            // edge_index[0], length E
  const int* dst = ei + N_EDGES;                  // edge_index[1], length E
  const float* W0 = (const float*)d_in[2]; const float* b0 = (const float*)d_in[3];
  const float* W1 = (const float*)d_in[4]; const float* b1 = (const float*)d_in[5];
  const float* W2 = (const float*)d_in[6]; const float* b2 = (const float*)d_in[7];
  const float* W3 = (const float*)d_in[8]; const float* b3 = (const float*)d_in[9];

  // workspace layout (floats): [dinv | bufH | bufA | bufX]  ~74 MB total
  float* ws   = (float*)d_ws;
  float* dinv = ws;
  float* bufH = ws + 65536;
  float* bufA = bufH + (size_t)N_NODES * HID;
  float* bufX = bufA + (size_t)N_NODES * HID;

  // normalization coefficients (depend only on edge_index -> compute once per launch)
  k_zero<<<(N_NODES + 255) / 256, 256, 0, stream>>>(dinv, N_NODES);
  k_deg <<<(N_EDGES + 255) / 256, 256, 0, stream>>>(dst, dinv, N_EDGES);
  k_dinv<<<(N_NODES + 255) / 256, 256, 0, stream>>>(dinv, N_NODES);

  run_layer<HID>      (x,    W0, b0, bufH, bufA, bufX,          dinv, src, dst, 0, stream);
  run_layer<HID>      (bufX, W1, b1, bufH, bufA, bufX,          dinv, src, dst, 0, stream);
  run_layer<HID>      (bufX, W2, b2, bufH, bufA, bufX,          dinv, src, dst, 0, stream);
  run_layer<N_CLASSES>(bufX, W3, b3, bufH, bufA, (float*)d_out, dinv, src, dst, 1, stream);
}